// Graph_8564164788735
// MI455X (gfx1250) — compile-verified
//
#include <hip/hip_runtime.h>
#include <hip/hip_bf16.h>

#define N_NODES 1024
#define NCLS    128
#define HID     256
#define CT      32           // classes per block
#define NTHREADS 512         // 16 waves (wave32)

typedef float v2f __attribute__((ext_vector_type(2)));
typedef float v8f __attribute__((ext_vector_type(8)));

__device__ __forceinline__ v8f wmma4(v2f a, v2f b, v8f c) {
  // D = A(16x4) * B(4x16) + C, fp32, wave32
  return __builtin_amdgcn_wmma_f32_16x16x4_f32(false, a, false, b, (short)0, c, false, false);
}

__device__ __forceinline__ float sigmoidf_(float v) { return 1.0f / (1.0f + __expf(-v)); }

#define VZERO ((v8f){0.f,0.f,0.f,0.f,0.f,0.f,0.f,0.f})

// ---------------------------------------------------------------------------
// Weight transpose + k-pair interleave so WMMA B-operands are single b64 loads.
// in : [rowsJ][colsK] row-major (W[j][k]);  out[((k>>1)*rowsJ + j)*2 + (k&1)] = W[j][k]
// ---------------------------------------------------------------------------
__global__ __launch_bounds__(256) void xpose_pair(const float* __restrict__ in,
                                                  float* __restrict__ out,
                                                  int rowsJ, int colsK) {
  int idx = blockIdx.x * 256 + threadIdx.x;
  if (idx < rowsJ * colsK) {
    int j = idx / colsK;
    int k = idx - j * colsK;
    out[((k >> 1) * rowsJ + j) * 2 + (k & 1)] = in[idx];
  }
}

// ---------------------------------------------------------------------------
// hs[c,h] = sum_n hidden[n,c,h]   (layer0: hidden[n,c,h] == x[n,h])
// ---------------------------------------------------------------------------
__global__ __launch_bounds__(256) void hs_kernel(const float* __restrict__ hin,
                                                 const float* __restrict__ x,
                                                 float* __restrict__ hs, int layer0) {
  int c = blockIdx.x, h = threadIdx.x;
  float s0 = 0.f, s1 = 0.f, s2 = 0.f, s3 = 0.f;
  if (layer0) {
    for (int nn = 0; nn < N_NODES; nn += 4) {
      s0 += x[(nn + 0) * HID + h];
      s1 += x[(nn + 1) * HID + h];
      s2 += x[(nn + 2) * HID + h];
      s3 += x[(nn + 3) * HID + h];
    }
  } else {
    const float* p = hin + (size_t)c * HID + h;
    const size_t stride = (size_t)NCLS * HID;
    for (int nn = 0; nn < N_NODES; nn += 4) {
      s0 += p[(nn + 0) * stride];
      s1 += p[(nn + 1) * stride];
      s2 += p[(nn + 2) * stride];
      s3 += p[(nn + 3) * stride];
    }
  }
  hs[c * HID + h] = (s0 + s1) + (s2 + s3);
}

// ---------------------------------------------------------------------------
// Fused per-(node, 32-class tile) layer kernel. All GEMMs via v_wmma_f32_16x16x4_f32.
// Wave w owns output-column tile nt=w (cols w*16..w*16+15) for BOTH row tiles,
// so every B-operand load is shared across the mt dimension.
// ---------------------------------------------------------------------------
__global__ __launch_bounds__(NTHREADS) void node_kernel(
    const float* __restrict__ hin, const float* __restrict__ x,
    const float* __restrict__ hs, const float* __restrict__ Kp,
    const float* __restrict__ w3sw, const float* __restrict__ w4sw,
    const float* __restrict__ w5sw, const float* __restrict__ u3sw,
    const float* __restrict__ u5sw,
    const float* __restrict__ b3, const float* __restrict__ bu3,
    const float* __restrict__ b4, const float* __restrict__ b5,
    const float* __restrict__ bu5,
    float* __restrict__ hout, int layer0) {
  // 160KB LDS: panel/q 32K | hslice 32K | left 32K | right 32K | KT 16K | KR 16K
  __shared__ float smem[40960];
  float* panel = smem;            // diff k-panel (pair-interleaved), later q (row-major)
  float* hsl   = smem + 8192;     // [32][256]
  float* lft   = smem + 16384;    // [32][256]
  float* rgt   = smem + 24576;    // [32][256]
  float* KT    = smem + 32768;    // [32][128]  KT[c][k] = K[k][c0+c]
  float* KR    = smem + 36864;    // [32][128]  KR[c][k] = K[c0+c][k]

  const int t    = threadIdx.x;
  const int wave = t >> 5;
  const int lane = t & 31;
  const int ln   = lane & 15;     // N (or M) index within tile
  const int lh   = lane >> 4;     // K-half selector
  const int n    = blockIdx.y;
  const int c0   = blockIdx.x * CT;

  const float* hrow = layer0 ? (x + (size_t)n * HID) : (hin + (size_t)n * NCLS * HID);

  // ---- phase 0: stage K tiles and hidden slice ----
  for (int idx = t; idx < CT * NCLS; idx += NTHREADS) {
    int k = idx >> 5, c = idx & 31;
    KT[c * NCLS + k] = Kp[k * NCLS + c0 + c];
  }
  for (int idx = t; idx < CT * NCLS; idx += NTHREADS) {
    int c = idx >> 7, k = idx & 127;
    KR[c * NCLS + k] = Kp[(c0 + c) * NCLS + k];
  }
  for (int idx = t; idx < CT * HID; idx += NTHREADS) {
    int c = idx >> 8, h = idx & 255;
    hsl[idx] = layer0 ? hrow[h] : hrow[(c0 + c) * HID + h];
  }

  // ---- phase 1: left = KT @ diff, right = KR @ diff (diff staged in k-panels) ----
  // jobs per wave (shared B): 0: KT/mt0, 1: KT/mt1, 2: KR/mt0, 3: KR/mt1, nt = wave
  v8f acc1[4];
#pragma unroll
  for (int j = 0; j < 4; ++j) acc1[j] = VZERO;

  for (int kp = 0; kp < 4; ++kp) {
    __syncthreads();
    for (int idx = t; idx < 32 * HID; idx += NTHREADS) {
      int k = idx >> 8, h = idx & 255;
      int kg = kp * 32 + k;
      float hv = layer0 ? hrow[h] : hrow[kg * HID + h];
      panel[((k >> 1) * HID + h) * 2 + (k & 1)] = hs[kg * HID + h] - hv;
    }
    __syncthreads();
    const float* bcol = panel + (wave * 16 + ln) * 2 + lh * 512;
    const float* a0 = KT + (ln) * NCLS + kp * 32 + 2 * lh;
    const float* a1 = KT + (16 + ln) * NCLS + kp * 32 + 2 * lh;
    const float* a2 = KR + (ln) * NCLS + kp * 32 + 2 * lh;
    const float* a3 = KR + (16 + ln) * NCLS + kp * 32 + 2 * lh;
#pragma unroll
    for (int kk = 0; kk < 32; kk += 4) {
      v2f b = *(const v2f*)(bcol + (kk >> 1) * 512);
      acc1[0] = wmma4(*(const v2f*)(a0 + kk), b, acc1[0]);
      acc1[1] = wmma4(*(const v2f*)(a1 + kk), b, acc1[1]);
      acc1[2] = wmma4(*(const v2f*)(a2 + kk), b, acc1[2]);
      acc1[3] = wmma4(*(const v2f*)(a3 + kk), b, acc1[3]);
    }
  }
#pragma unroll
  for (int j = 0; j < 4; ++j) {
    float* O = (j >= 2) ? rgt : lft;
    int mt = j & 1;
#pragma unroll
    for (int r = 0; r < 8; ++r)
      O[(mt * 16 + r + 8 * lh) * HID + wave * 16 + ln] = acc1[j][r];
  }
  __syncthreads();

  // ---- phase 2: zpre/rpre/hpre = [left|right|h] x {W3,W4,W5,U3}; q -> panel ----
  const int col = wave * 16 + ln;
  const float* bz = w3sw + col * 2 + lh * 512;
  const float* br = w4sw + col * 2 + lh * 512;
  const float* bh = w5sw + col * 2 + lh * 512;
  const float* bu = u3sw + col * 2 + lh * 512;
  const float* aL[2] = { lft + ln * HID + 2 * lh, lft + (16 + ln) * HID + 2 * lh };
  const float* aR[2] = { rgt + ln * HID + 2 * lh, rgt + (16 + ln) * HID + 2 * lh };
  const float* aH[2] = { hsl + ln * HID + 2 * lh, hsl + (16 + ln) * HID + 2 * lh };

  v8f az[2] = {VZERO, VZERO}, ar[2] = {VZERO, VZERO};
  v8f ah[2] = {VZERO, VZERO}, au[2] = {VZERO, VZERO};
  for (int kk = 0; kk < HID; kk += 4) {
    int o = (kk >> 1) * 512;
    v2f bz1 = *(const v2f*)(bz + o);
    v2f bz2 = *(const v2f*)(bz + o + 65536);   // rows 256..511 of W3
    v2f br1 = *(const v2f*)(br + o);
    v2f br2 = *(const v2f*)(br + o + 65536);
    v2f bh1 = *(const v2f*)(bh + o);
    v2f bh2 = *(const v2f*)(bh + o + 65536);
    v2f bu1 = *(const v2f*)(bu + o);
#pragma unroll
    for (int mt = 0; mt < 2; ++mt) {
      v2f xa = *(const v2f*)(aL[mt] + kk);
      v2f xb = *(const v2f*)(aR[mt] + kk);
      v2f xh = *(const v2f*)(aH[mt] + kk);
      az[mt] = wmma4(xa, bz1, az[mt]); az[mt] = wmma4(xb, bz2, az[mt]);
      ar[mt] = wmma4(xa, br1, ar[mt]); ar[mt] = wmma4(xb, br2, ar[mt]);
      ah[mt] = wmma4(xa, bh1, ah[mt]); ah[mt] = wmma4(xb, bh2, ah[mt]);
      au[mt] = wmma4(xh, bu1, au[mt]);
    }
  }
  float cz = b3[col] + bu3[col];
  float cr = b4[col] + bu3[col];
  float ch = b5[col] + bu5[col];
  float zk[2][8], hk[2][8];
#pragma unroll
  for (int mt = 0; mt < 2; ++mt) {
#pragma unroll
    for (int r = 0; r < 8; ++r) {
      int m = mt * 16 + r + 8 * lh;
      zk[mt][r] = sigmoidf_(az[mt][r] + au[mt][r] + cz);      // zv
      hk[mt][r] = ah[mt][r] + ch;                             // hpre (pre-U5, pre-tanh)
      float rv = sigmoidf_(ar[mt][r] + au[mt][r] + cr);
      panel[m * HID + col] = rv * hsl[m * HID + col];         // q (row-major, A-operand)
    }
  }
  __syncthreads();

  // ---- phase 4: hv = tanh(hpre + q @ U5^T); h' = (1-zv)*h + zv*hv ----
  const float* aQ[2] = { panel + ln * HID + 2 * lh, panel + (16 + ln) * HID + 2 * lh };
  const float* b5u = u5sw + col * 2 + lh * 512;
  v8f acc[2] = {VZERO, VZERO};
  for (int kk = 0; kk < HID; kk += 4) {
    v2f b = *(const v2f*)(b5u + (kk >> 1) * 512);
    acc[0] = wmma4(*(const v2f*)(aQ[0] + kk), b, acc[0]);
    acc[1] = wmma4(*(const v2f*)(aQ[1] + kk), b, acc[1]);
  }
#pragma unroll
  for (int mt = 0; mt < 2; ++mt) {
#pragma unroll
    for (int r = 0; r < 8; ++r) {
      int m = mt * 16 + r + 8 * lh;
      float hold = hsl[m * HID + col];
      float hv = tanhf(hk[mt][r] + acc[mt][r]);
      float z  = zk[mt][r];
      hout[((size_t)(n * NCLS + c0 + m)) * HID + col] = (1.0f - z) * hold + z * hv;
    }
  }
}

// ---------------------------------------------------------------------------
extern "C" void kernel_launch(void* const* d_in, const int* in_sizes, int n_in,
                              void* d_out, int out_size, void* d_ws, size_t ws_size,
                              hipStream_t stream) {
  const float* x   = (const float*)d_in[0];
  const float* Kp  = (const float*)d_in[1];
  const float* w3w = (const float*)d_in[2];
  const float* b3w = (const float*)d_in[3];
  const float* u3  = (const float*)d_in[4];
  const float* bu3 = (const float*)d_in[5];
  const float* w4w = (const float*)d_in[6];
  const float* b4w = (const float*)d_in[7];
  const float* w5w = (const float*)d_in[8];
  const float* b5w = (const float*)d_in[9];
  const float* u5  = (const float*)d_in[10];
  const float* bu5 = (const float*)d_in[11];
  float* out = (float*)d_out;

  float* ws     = (float*)d_ws;
  float* hidMid = ws;                                      // 128 MB ping buffer
  float* hsbuf  = hidMid + (size_t)N_NODES * NCLS * HID;
  float* w3sw   = hsbuf + NCLS * HID;
  float* w4sw   = w3sw + 2 * HID * HID;
  float* w5sw   = w4sw + 2 * HID * HID;
  float* u3sw   = w5sw + 2 * HID * HID;
  float* u5sw   = u3sw + HID * HID;

  // Pre-swizzle weights (tiny; L2-resident thereafter)
  xpose_pair<<<dim3((HID * 2 * HID + 255) / 256), 256, 0, stream>>>(w3w, w3sw, HID, 2 * HID);
  xpose_pair<<<dim3((HID * 2 * HID + 255) / 256), 256, 0, stream>>>(w4w, w4sw, HID, 2 * HID);
  xpose_pair<<<dim3((HID * 2 * HID + 255) / 256), 256, 0, stream>>>(w5w, w5sw, HID, 2 * HID);
  xpose_pair<<<dim3((HID * HID + 255) / 256), 256, 0, stream>>>(u3, u3sw, HID, HID);
  xpose_pair<<<dim3((HID * HID + 255) / 256), 256, 0, stream>>>(u5, u5sw, HID, HID);

  dim3 grid(NCLS / CT, N_NODES);

  // layer 0: hidden0[n,k,h] == x[n,h] (broadcast handled in-kernel) -> hidMid
  hs_kernel<<<NCLS, HID, 0, stream>>>(nullptr, x, hsbuf, 1);
  node_kernel<<<grid, NTHREADS, 0, stream>>>(nullptr, x, hsbuf, Kp, w3sw, w4sw, w5sw,
                                             u3sw, u5sw, b3w, bu3, b4w, b5w, bu5, hidMid, 1);
  // layer 1: hidMid -> d_out
  hs_kernel<<<NCLS, HID, 0, stream>>>(hidMid, x, hsbuf, 0);
  node_kernel<<<grid, NTHREADS, 0, stream>>>(hidMid, x, hsbuf, Kp, w3sw, w4sw, w5sw,
                                             u3sw, u5sw, b3w, bu3, b4w, b5w, bu5, out, 0);
}